// FocalEMA_89756226551855
// MI455X (gfx1250) — compile-verified
//
#include <hip/hip_runtime.h>
#include <hip/hip_bf16.h>

#define NCLS 128
#define NSLOT 64
#define IGNORE_INDEX (-100)

typedef __attribute__((ext_vector_type(2))) float v2f;
typedef __attribute__((ext_vector_type(8))) float v8f;

// ---------------------------------------------------------------------------
// Phase 1: one wave32 per row of 128 fp32 logits.
//  - lane l holds columns 4l..4l+3 via one float4 load (512B/row, coalesced)
//  - fused: argmax + row max + sum(exp(x-m)) + x[gt] in a single pass
//  - confusion scatter: fp32 atomic into 128x128 bins
//  - CE scatter: slotted (128 x 64) fp32 atomics to bound per-address contention
// ---------------------------------------------------------------------------
__global__ void __launch_bounds__(256) focal_phase1(
    const float* __restrict__ lg, const int* __restrict__ gt,
    float* __restrict__ conf, float* __restrict__ ceslot,
    int bs, int iters) {
  const int lane  = threadIdx.x & 31;
  const int warp  = threadIdx.x >> 5;
  const int gwave = blockIdx.x * 8 + warp;
  const int total_waves = gridDim.x * 8;
  const float4* lg4 = (const float4*)lg;

  for (int it = 0; it < iters; ++it) {
    int row = it * total_waves + gwave;
    if (row >= bs) break;

    // CDNA5 prefetch for the next grid-stride iteration (global_prefetch_b8)
    int nrow = row + total_waves;
    if (it + 1 < iters && nrow < bs)
      __builtin_prefetch(&lg4[(size_t)nrow * 32 + lane], 0, 1);

    float4 v = lg4[(size_t)row * 32 + lane];

    // local argmax, first-max-wins
    float m = v.x; int sub = 0;
    if (v.y > m) { m = v.y; sub = 1; }
    if (v.z > m) { m = v.z; sub = 2; }
    if (v.w > m) { m = v.w; sub = 3; }
    int col = lane * 4 + sub;

    // wave32 butterfly argmax (ties -> lower column index)
    #pragma unroll
    for (int off = 16; off >= 1; off >>= 1) {
      float om = __shfl_xor(m, off, 32);
      int   oc = __shfl_xor(col, off, 32);
      if (om > m || (om == m && oc < col)) { m = om; col = oc; }
    }

    // sum of exp(x - m)
    float s = __expf(v.x - m) + __expf(v.y - m) + __expf(v.z - m) + __expf(v.w - m);
    #pragma unroll
    for (int off = 16; off >= 1; off >>= 1) s += __shfl_xor(s, off, 32);

    int g      = gt[row];
    bool valid = (g != IGNORE_INDEX);
    int sg     = valid ? g : 0;

    // extract logit at column sg (wave-uniform index)
    int   gsub = sg & 3;
    float cand = (gsub == 0) ? v.x : (gsub == 1) ? v.y : (gsub == 2) ? v.z : v.w;
    float xg   = __shfl(cand, sg >> 2, 32);

    float ce = m + __logf(s) - xg;  // -log_softmax[gt]

    if (lane == 0 && valid) {
      atomicAdd(&conf[sg * NCLS + col], 1.0f);
      atomicAdd(&ceslot[sg * NSLOT + (gwave & (NSLOT - 1))], ce);
    }
  }
}

// ---------------------------------------------------------------------------
// Phase 2: single block, 256 threads (8 waves).
//  - EMA row sums via V_WMMA_F32_16X16X4_F32: rowsum(ema) = ema(128x128) @ ones.
//    Wave w owns rows 16w..16w+15; A tile = 0.8*conf + 0.2*ema_conf, B = ones.
//  - mispred -> max -> clipped class weights -> weighted CE sum -> scalar loss
// ---------------------------------------------------------------------------
__global__ void __launch_bounds__(256) focal_phase2(
    const float* __restrict__ conf, const float* __restrict__ emac,
    const float* __restrict__ ceslot, float* __restrict__ out, float inv_bs) {
  __shared__ float rs[NCLS];
  __shared__ float mis[NCLS];
  __shared__ float red[NCLS];

  const int tid   = threadIdx.x;
  const int lane  = tid & 31;
  const int warp  = tid >> 5;
  const int lm    = lane & 15;
  const int hi    = lane >> 4;      // 0: K pair {0,1}, 1: K pair {2,3}
  const int tileM = warp * 16;
  const int row   = tileM + lm;

  // All 256 threads execute this uniformly (EXEC all-1s for WMMA).
  v8f acc = {0.f, 0.f, 0.f, 0.f, 0.f, 0.f, 0.f, 0.f};
  v2f b;  b[0] = 1.0f; b[1] = 1.0f;   // ones vector, replicated
  for (int k = 0; k < NCLS; k += 4) {
    int cb = k + hi * 2;
    float c0 = conf[row * NCLS + cb];
    float c1 = conf[row * NCLS + cb + 1];
    float e0 = emac[row * NCLS + cb];
    float e1 = emac[row * NCLS + cb + 1];
    v2f a;  a[0] = 0.8f * c0 + 0.2f * e0;  a[1] = 0.8f * c1 + 0.2f * e1;
    // D = A(16x4 f32) * B(4x16 ones) + C  -> every column of D = row sum
    acc = __builtin_amdgcn_wmma_f32_16x16x4_f32(
        /*neg_a=*/false, a, /*neg_b=*/false, b,
        /*c_mod=*/(short)0, acc, /*reuse_a=*/false, /*reuse_b=*/false);
  }
  // N=0 column lives in lane 0 (M = tileM+v) and lane 16 (M = tileM+8+v)
  if (lm == 0) {
    #pragma unroll
    for (int vv = 0; vv < 8; ++vv) rs[tileM + hi * 8 + vv] = acc[vv];
  }
  __syncthreads();

  if (tid < NCLS) {
    float d = 0.8f * conf[tid * NCLS + tid] + 0.2f * emac[tid * NCLS + tid];
    mis[tid] = rs[tid] - d;
    red[tid] = mis[tid];
  }
  __syncthreads();
  for (int s2 = 64; s2 > 0; s2 >>= 1) {
    if (tid < s2) red[tid] = fmaxf(red[tid], red[tid + s2]);
    __syncthreads();
  }
  float maxm = red[0];
  __syncthreads();

  if (tid < NCLS) {
    float cw = fminf(1.2f, maxm / (mis[tid] + 1e-6f));
    float part = 0.f;
    #pragma unroll 8
    for (int s = 0; s < NSLOT; ++s) part += ceslot[tid * NSLOT + s];
    red[tid] = cw * part;
  }
  __syncthreads();
  for (int s2 = 64; s2 > 0; s2 >>= 1) {
    if (tid < s2) red[tid] += red[tid + s2];
    __syncthreads();
  }
  if (tid == 0) out[0] = red[0] * inv_bs;
}

extern "C" void kernel_launch(void* const* d_in, const int* in_sizes, int n_in,
                              void* d_out, int out_size, void* d_ws, size_t ws_size,
                              hipStream_t stream) {
  const float* lg   = (const float*)d_in[0];
  const int*   gt   = (const int*)d_in[1];
  const float* emac = (const float*)d_in[2];
  float*       out  = (float*)d_out;

  const int bs = in_sizes[0] / NCLS;

  float* conf   = (float*)d_ws;             // 128*128 fp32 = 64 KB
  float* ceslot = conf + NCLS * NCLS;       // 128*64  fp32 = 32 KB

  hipMemsetAsync(d_ws, 0, (size_t)(NCLS * NCLS + NCLS * NSLOT) * sizeof(float),
                 stream);

  const int iters = 4;
  int waves_needed = (bs + iters - 1) / iters;
  int nblocks = (waves_needed + 7) / 8;
  focal_phase1<<<nblocks, 256, 0, stream>>>(lg, gt, conf, ceslot, bs, iters);
  focal_phase2<<<1, 256, 0, stream>>>(conf, emac, ceslot, out, 1.0f / (float)bs);
}